// PhaGatModel_29721173688775
// MI455X (gfx1250) — compile-verified
//
#include <hip/hip_runtime.h>

typedef float v2f __attribute__((ext_vector_type(2)));
typedef float v8f __attribute__((ext_vector_type(8)));

// C[M x 64] = A[M x K] @ W[K x 64] + bias, fp32, via V_WMMA_F32_16X16X4_F32.
// Block = 256 threads = 8 waves. Wave w: column tile n0=(w&3)*16, row tiles
// (blockIdx.x*2 + (w>>2)) with grid-stride. B fragments (this wave's 16
// columns of W) are preloaded into VGPRs once and reused over all row tiles.
// A loads are unpredicated (row index clamped, so always in-bounds); garbage
// from clamped rows only reaches output rows >= Mrows, whose stores are
// guarded. Loads are issued as a batch before the WMMA chain for MLP.
template <int K>
__global__ __launch_bounds__(256) void gemm_bias_wmma(const float* __restrict__ A,
                                                      const float* __restrict__ W,
                                                      const float* __restrict__ bias,
                                                      float* __restrict__ C,
                                                      int Mrows) {
    constexpr int NK = K / 4;                 // k-steps of 4
    const int lane = threadIdx.x & 31;
    const int wave = threadIdx.x >> 5;
    const int half = lane >> 4;               // 0: lanes 0-15, 1: lanes 16-31
    const int col  = lane & 15;               // N (and A's M) index within tile
    const int n0   = (wave & 3) * 16;         // output column tile base

    // B-matrix fragments: VGPR v, lane L holds W[k0 + v + 2*(L/16)][n0 + L%16]
    v2f bfrag[NK];
#pragma unroll
    for (int s = 0; s < NK; ++s) {
        const int k = 4 * s + 2 * half;
        bfrag[s].x = W[(k + 0) * 64 + n0 + col];
        bfrag[s].y = W[(k + 1) * 64 + n0 + col];
    }
    const float bv = bias[n0 + col];

    const int numTiles = (Mrows + 15) >> 4;
    for (int tile = blockIdx.x * 2 + (wave >> 2); tile < numTiles; tile += gridDim.x * 2) {
        const int row  = tile * 16 + col;     // A-matrix M index held by this lane
        const int rowc = row < Mrows ? row : Mrows - 1;   // clamped: load always legal
        const float* __restrict__ abase = A + (long long)rowc * K + 2 * half;

        // Batch-issue all A-fragment loads (b64 each), then consume.
        v2f afrag[NK];
#pragma unroll
        for (int s = 0; s < NK; ++s)
            afrag[s] = *(const v2f*)(abase + 4 * s);

        v8f acc = {0.f, 0.f, 0.f, 0.f, 0.f, 0.f, 0.f, 0.f};
#pragma unroll
        for (int s = 0; s < NK; ++s)
            acc = __builtin_amdgcn_wmma_f32_16x16x4_f32(
                /*neg_a=*/false, afrag[s], /*neg_b=*/false, bfrag[s],
                /*c_mod=*/(short)0, acc, /*reuse_a=*/false, /*reuse_b=*/false);

        // D layout: VGPR v, lane L -> C[tile*16 + v + 8*(L/16)][n0 + L%16]
        float* __restrict__ cbase = C + (long long)(tile * 16 + 8 * half) * 64 + n0 + col;
        if (tile * 16 + 16 <= Mrows) {        // full tile: unguarded stores
#pragma unroll
            for (int v = 0; v < 8; ++v)
                cbase[(long long)v * 64] = acc[v] + bv;
        } else {                              // tail tile: per-row guard
#pragma unroll
            for (int v = 0; v < 8; ++v) {
                const int r = tile * 16 + v + 8 * half;
                if (r < Mrows) cbase[(long long)v * 64] = acc[v] + bv;
            }
        }
    }
}

// out[row][0:64] = (env[row]==0) ? 0 : emb[env[row]-1][0:64]
// One float4 per thread, 16 threads per 256-byte row (coalesced writes;
// gathered reads hit L2 since emb (128 MB) fits in the 192 MB L2).
__global__ __launch_bounds__(256) void gather_pairs(const float* __restrict__ emb,
                                                    const int* __restrict__ env,
                                                    float* __restrict__ out,
                                                    long long totalRows) {
    const long long gid = (long long)blockIdx.x * blockDim.x + threadIdx.x;
    const long long row = gid >> 4;
    const int c4 = (int)(gid & 15);
    if (row >= totalRows) return;
    const int idx = env[row];
    float4 val = {0.f, 0.f, 0.f, 0.f};
    if (idx > 0)
        val = ((const float4*)(emb + (long long)(idx - 1) * 64))[c4];
    ((float4*)(out + row * 64))[c4] = val;
}

extern "C" void kernel_launch(void* const* d_in, const int* in_sizes, int n_in,
                              void* d_out, int out_size, void* d_ws, size_t ws_size,
                              hipStream_t stream) {
    (void)n_in; (void)out_size; (void)ws_size;
    const float* target_features    = (const float*)d_in[0];  // [N, 40]
    const float* feature_dist_graph = (const float*)d_in[1];  // [E, 64]
    const float* W_emb              = (const float*)d_in[2];  // [40, 64]
    const float* b_emb              = (const float*)d_in[3];  // [64]
    const float* W_dist             = (const float*)d_in[4];  // [64, 64]
    const float* b_dist             = (const float*)d_in[5];  // [64]
    const int*   start_end_env      = (const int*)d_in[6];    // [E, 2]

    const int N = in_sizes[0] / 40;
    const int E = in_sizes[1] / 64;

    float* emb_ws   = (float*)d_ws;                           // [N, 64] scratch
    float* out_gat  = (float*)d_out;                          // [2E, 64]
    float* out_dist = (float*)d_out + (size_t)2 * E * 64;     // [E, 64]

    auto blocksFor = [](int rows) {
        const int tiles  = (rows + 15) / 16;
        const int blocks = (tiles + 1) / 2;   // 2 row-tiles per block per pass
        return blocks < 4096 ? blocks : 4096; // grid-stride covers the rest
    };

    // 1) emb = target_features @ W_emb + b_emb  -> workspace
    gemm_bias_wmma<40><<<blocksFor(N), 256, 0, stream>>>(
        target_features, W_emb, b_emb, emb_ws, N);

    // 2) dist_emb = feature_dist_graph @ W_dist + b_dist -> second output
    gemm_bias_wmma<64><<<blocksFor(E), 256, 0, stream>>>(
        feature_dist_graph, W_dist, b_dist, out_dist, E);

    // 3) gather padded[start_end_env] -> first output (stream order gives the
    //    emb -> gather dependency)
    const long long rows    = (long long)2 * E;
    const long long threads = rows * 16;
    const int gblocks = (int)((threads + 255) / 256);
    gather_pairs<<<gblocks, 256, 0, stream>>>(emb_ws, start_end_env, out_gat, rows);
}